// RGCNLayer_27006754357409
// MI455X (gfx1250) — compile-verified
//
#include <hip/hip_runtime.h>

#define N_NODES  100000
#define N_EDGES  3200000
#define IN_FEAT  64
#define OUT_FEAT 64
#define NUM_RELS 16
#define NTILES   (N_NODES / 16)   // 6250, exact

typedef __attribute__((ext_vector_type(2))) float v2f;
typedef __attribute__((ext_vector_type(4))) float v4f;
typedef __attribute__((ext_vector_type(8))) float v8f;

// ---------------------------------------------------------------------------
// Zero the output accumulator.
// ---------------------------------------------------------------------------
__global__ void rgcn_zero(float* __restrict__ out, int n) {
    int i = blockIdx.x * blockDim.x + threadIdx.x;
    if (i < n) out[i] = 0.0f;
}

// ---------------------------------------------------------------------------
// Transform kernel: tr[n, rc, o] = sum_d h[n,d] * weight[r0+rc, d, o]
// One wave32 computes a 16(node) x 64(out) tile for one relation using
// V_WMMA_F32_16X16X4_F32. K-loop: 64/4 = 16 steps, 4 accumulators (col tiles).
//
// f32 WMMA VGPR layouts (ISA 7.12.2):
//   A (16x4):  lane L -> M = L%16; VGPR v holds K = (L/16)*2 + v
//   B (4x16):  lane L -> N = L%16; VGPR v holds K = (L/16)*2 + v
//   C/D (16x16): lane L -> N = L%16; VGPR j holds M = j + 8*(L/16)
// ---------------------------------------------------------------------------
__global__ __launch_bounds__(32) void rgcn_transform(
    const float* __restrict__ h,      // [N_NODES, IN_FEAT]
    const float* __restrict__ w,      // [NUM_RELS, IN_FEAT, OUT_FEAT]
    float* __restrict__ tr,           // [N_NODES, nrel, OUT_FEAT]
    int r0, int nrel) {
    const int b     = blockIdx.x;
    const int ntile = b % NTILES;
    const int rc    = b / NTILES;
    const int r     = r0 + rc;

    const int lane = threadIdx.x;
    const int half = lane >> 4;       // 0 or 1
    const int m    = lane & 15;

    const float* __restrict__ hrow = h + (size_t)(ntile * 16 + m) * IN_FEAT;
    const float* __restrict__ wr   = w + (size_t)r * IN_FEAT * OUT_FEAT;

    v8f acc0 = {}, acc1 = {}, acc2 = {}, acc3 = {};

#pragma unroll
    for (int k0 = 0; k0 < IN_FEAT; k0 += 4) {
        const int ka = k0 + half * 2;                 // this lane's K base
        // A fragment: two consecutive K values of this lane's node row.
        v2f a = *(const v2f*)(hrow + ka);
        // B fragments: weight rows ka and ka+1, 16 columns per tile.
        const float* __restrict__ wk = wr + (size_t)ka * OUT_FEAT;
        v2f b0, b1, b2, b3;
        b0.x = wk[m];           b0.y = wk[OUT_FEAT + m];
        b1.x = wk[16 + m];      b1.y = wk[OUT_FEAT + 16 + m];
        b2.x = wk[32 + m];      b2.y = wk[OUT_FEAT + 32 + m];
        b3.x = wk[48 + m];      b3.y = wk[OUT_FEAT + 48 + m];

        acc0 = __builtin_amdgcn_wmma_f32_16x16x4_f32(false, a, false, b0,
                                                     (short)0, acc0, false, false);
        acc1 = __builtin_amdgcn_wmma_f32_16x16x4_f32(false, a, false, b1,
                                                     (short)0, acc1, false, false);
        acc2 = __builtin_amdgcn_wmma_f32_16x16x4_f32(false, a, false, b2,
                                                     (short)0, acc2, false, false);
        acc3 = __builtin_amdgcn_wmma_f32_16x16x4_f32(false, a, false, b3,
                                                     (short)0, acc3, false, false);
    }

    // Store C/D tiles: VGPR j holds row M = j + 8*half, column N = m.
    const size_t row_stride = (size_t)nrel * OUT_FEAT;
    float* __restrict__ o =
        tr + (size_t)(ntile * 16) * row_stride + (size_t)rc * OUT_FEAT;
#pragma unroll
    for (int j = 0; j < 8; ++j) {
        float* __restrict__ orow = o + (size_t)(j + half * 8) * row_stride;
        orow[m]      = acc0[j];
        orow[16 + m] = acc1[j];
        orow[32 + m] = acc2[j];
        orow[48 + m] = acc3[j];
    }
}

// ---------------------------------------------------------------------------
// Edge scatter: 16 threads per edge, 4 consecutive floats per thread.
// out[dst] += tr[src, rel-r0, :] * norm   via f32 atomics (hit L2 atomic units;
// the 25.6MB output fits in the 192MB L2).
// ---------------------------------------------------------------------------
__global__ void rgcn_edge(
    const float* __restrict__ tr,     // [N_NODES, nrel, OUT_FEAT]
    const float* __restrict__ norm,   // [N_EDGES]
    const int*   __restrict__ src,
    const int*   __restrict__ dst,
    const int*   __restrict__ rel,
    float* __restrict__ out,          // [N_NODES, OUT_FEAT]
    int r0, int nrel) {
    const long long gid = (long long)blockIdx.x * blockDim.x + threadIdx.x;
    const int e = (int)(gid >> 4);
    if (e >= N_EDGES) return;
    const int r = rel[e] - r0;
    if (r < 0 || r >= nrel) return;

    const int f = (int)(gid & 15) * 4;
    const float nr = norm[e];
    const v4f v = *(const v4f*)(tr + (size_t)src[e] * ((size_t)nrel * OUT_FEAT)
                                   + (size_t)r * OUT_FEAT + f);
    float* __restrict__ o = out + (size_t)dst[e] * OUT_FEAT + f;
    atomicAdd(o + 0, v.x * nr);
    atomicAdd(o + 1, v.y * nr);
    atomicAdd(o + 2, v.z * nr);
    atomicAdd(o + 3, v.w * nr);
}

// ---------------------------------------------------------------------------
// Fallback (no workspace): direct per-(edge,feature) GEMV + atomic scatter.
// ---------------------------------------------------------------------------
__global__ void rgcn_edge_direct(
    const float* __restrict__ h,
    const float* __restrict__ w,
    const float* __restrict__ norm,
    const int*   __restrict__ src,
    const int*   __restrict__ dst,
    const int*   __restrict__ rel,
    float* __restrict__ out) {
    const long long gid = (long long)blockIdx.x * blockDim.x + threadIdx.x;
    const int e = (int)(gid >> 6);
    if (e >= N_EDGES) return;
    const int f = (int)(gid & 63);
    const float* __restrict__ hr = h + (size_t)src[e] * IN_FEAT;
    const float* __restrict__ wr = w + (size_t)rel[e] * IN_FEAT * OUT_FEAT + f;
    float s = 0.0f;
#pragma unroll 8
    for (int d = 0; d < IN_FEAT; ++d) s += hr[d] * wr[(size_t)d * OUT_FEAT];
    atomicAdd(out + (size_t)dst[e] * OUT_FEAT + f, s * norm[e]);
}

// ---------------------------------------------------------------------------
// Final ReLU, in place.
// ---------------------------------------------------------------------------
__global__ void rgcn_relu(float* __restrict__ out, int n) {
    int i = blockIdx.x * blockDim.x + threadIdx.x;
    if (i < n) {
        float v = out[i];
        out[i] = v > 0.0f ? v : 0.0f;
    }
}

// ---------------------------------------------------------------------------
extern "C" void kernel_launch(void* const* d_in, const int* in_sizes, int n_in,
                              void* d_out, int out_size, void* d_ws, size_t ws_size,
                              hipStream_t stream) {
    const float* h    = (const float*)d_in[0];
    const float* w    = (const float*)d_in[1];
    const float* norm = (const float*)d_in[2];
    const int*   src  = (const int*)d_in[3];
    const int*   dst  = (const int*)d_in[4];
    const int*   rel  = (const int*)d_in[5];
    float*       out  = (float*)d_out;

    const int n_out = N_NODES * OUT_FEAT;                 // 6.4M
    rgcn_zero<<<(n_out + 255) / 256, 256, 0, stream>>>(out, n_out);

    // Pick the largest relation chunk whose transformed tensor fits in d_ws.
    int nrel = NUM_RELS;
    while (nrel > 1 &&
           (size_t)N_NODES * (size_t)nrel * OUT_FEAT * sizeof(float) > ws_size)
        nrel >>= 1;

    const bool use_ws =
        (size_t)N_NODES * (size_t)nrel * OUT_FEAT * sizeof(float) <= ws_size;

    if (use_ws) {
        float* tr = (float*)d_ws;
        const long long edge_threads = (long long)N_EDGES * 16;
        const int edge_grid = (int)((edge_threads + 255) / 256);
        for (int r0 = 0; r0 < NUM_RELS; r0 += nrel) {
            rgcn_transform<<<NTILES * nrel, 32, 0, stream>>>(h, w, tr, r0, nrel);
            rgcn_edge<<<edge_grid, 256, 0, stream>>>(tr, norm, src, dst, rel,
                                                     out, r0, nrel);
        }
    } else {
        const long long edge_threads = (long long)N_EDGES * 64;
        const int edge_grid = (int)((edge_threads + 255) / 256);
        rgcn_edge_direct<<<edge_grid, 256, 0, stream>>>(h, w, norm, src, dst,
                                                        rel, out);
    }

    rgcn_relu<<<(n_out + 255) / 256, 256, 0, stream>>>(out, n_out);
}